// IntraAgg_70179765617350
// MI455X (gfx1250) — compile-verified
//
#include <hip/hip_runtime.h>

// Problem constants (match the reference): N=50000, D=32, F=128, E=128, B=8192
#define NN 50000
#define DD 32
#define FF 128
#define EE 128
#define BB 8192

// LDS row stride (floats): 132 keeps float4 (16B) alignment per row
// (132*4 = 528 bytes, multiple of 16) and spreads A-fragment reads across
// banks (m*132 % 64 = m*4 -> 16 distinct banks across the half-wave).
#define LSTR 132

typedef float v2f __attribute__((ext_vector_type(2)));
typedef float v8f __attribute__((ext_vector_type(8)));

__global__ void zero_f32_kernel(float* __restrict__ p, int n) {
  int i = blockIdx.x * blockDim.x + threadIdx.x;
  if (i < n) p[i] = 0.0f;
}

__global__ void indeg_kernel(const int* __restrict__ adj, float* __restrict__ indeg) {
  int i = blockIdx.x * blockDim.x + threadIdx.x;
  if (i < NN * DD) {
    atomicAdd(&indeg[adj[i]], 1.0f);
  }
}

// One wave computes one 16x16 tile of  relu(A[16x128] @ W[128x128])  at column
// tile = wave index (E=128 -> 8 col tiles -> 8 waves per block).
// A is staged in LDS with row stride LSTR.
// V_WMMA_F32_16X16X4_F32 operand layouts (ISA 7.12.2):
//   A (16x4):  lane L -> row M=L%16;  VGPR0=K=(L<16?0:2), VGPR1=K+1  => contiguous v2f
//   B (4x16):  lane L -> col N=L%16;  VGPR0=row K=(L<16?0:2), VGPR1=K+1
//   C/D:       lane L, VGPR v -> M = v + (L<16?0:8), N = L%16
__device__ __forceinline__ void tile_gemm_relu_store(const float* __restrict__ W,
                                                     const float* lds,
                                                     float* __restrict__ out_tile) {
  const int wave = threadIdx.x >> 5;
  const int lane = threadIdx.x & 31;
  const int colBase = wave * 16;
  const int m    = lane & 15;   // row (A) / col (B,D)
  const int hi   = lane >> 4;   // 0 for lanes 0-15, 1 for lanes 16-31
  const int koff = hi << 1;     // 0 or 2

  v8f acc = {};
#pragma unroll
  for (int k0 = 0; k0 < FF; k0 += 4) {
    v2f a = *(const v2f*)(lds + m * LSTR + k0 + koff);
    v2f b;
    b.x = W[(k0 + koff) * EE + colBase + m];
    b.y = W[(k0 + koff + 1) * EE + colBase + m];
    acc = __builtin_amdgcn_wmma_f32_16x16x4_f32(
        /*neg_a=*/false, a, /*neg_b=*/false, b,
        /*c_mod=*/(short)0, acc, /*reuse_a=*/false, /*reuse_b=*/false);
  }

  const int rofs = hi << 3;  // 0 or 8
#pragma unroll
  for (int v = 0; v < 8; ++v) {
    out_tile[(size_t)(v + rofs) * EE + colBase + m] = fmaxf(acc[v], 0.0f);
  }
}

// Block 1 fused: mean over D=32 gathered neighbor rows -> LDS -> WMMA GEMM -> relu.
// grid.x = B/16 ; block = 256 (8 waves). Each wave aggregates 2 of the 16 rows.
__global__ void __launch_bounds__(256, 2)
batch_agg_gemm_kernel(const float* __restrict__ feat, const int* __restrict__ adj,
                      const int* __restrict__ nodes, const float* __restrict__ W,
                      float* __restrict__ out) {
  __shared__ float lds[16 * LSTR];
  const int wave = threadIdx.x >> 5;
  const int lane = threadIdx.x & 31;
  const int rowBase = blockIdx.x * 16;

#pragma unroll
  for (int rr = 0; rr < 2; ++rr) {
    const int r = wave * 2 + rr;
    const int node = nodes[rowBase + r];               // scalar per wave
    const int* nb = adj + (size_t)node * DD;
    float4 acc = make_float4(0.f, 0.f, 0.f, 0.f);
    for (int j = 0; j < DD; ++j) {
      const int nid = nb[j];                           // uniform -> SMEM load
      const float4 v = *(const float4*)(feat + (size_t)nid * FF + lane * 4);
      acc.x += v.x; acc.y += v.y; acc.z += v.z; acc.w += v.w;
    }
    const float s = 1.0f / (float)DD;
    *(float4*)(lds + r * LSTR + lane * 4) =
        make_float4(acc.x * s, acc.y * s, acc.z * s, acc.w * s);
  }
  __syncthreads();
  tile_gemm_relu_store(W, lds, out + (size_t)rowBase * EE);
}

// Block 2 fused: symmetric-normalized aggregation over all N nodes.
// scaled[nid] = feat[nid] * rsqrt(indeg[nid]); sum over D neighbors; / sqrt(D).
// grid.x = N/16 ; output rows offset by B in d_out (tuple concat order).
__global__ void __launch_bounds__(256, 2)
neigh_agg_gemm_kernel(const float* __restrict__ feat, const int* __restrict__ adj,
                      const float* __restrict__ indeg, const float* __restrict__ W,
                      float* __restrict__ out) {
  __shared__ float lds[16 * LSTR];
  const int wave = threadIdx.x >> 5;
  const int lane = threadIdx.x & 31;
  const int rowBase = blockIdx.x * 16;
  const float invSqrtD = 0.17677669529663688f;  // 1/sqrt(32)

#pragma unroll
  for (int rr = 0; rr < 2; ++rr) {
    const int r = wave * 2 + rr;
    const int node = rowBase + r;
    const int* nb = adj + (size_t)node * DD;
    float4 acc = make_float4(0.f, 0.f, 0.f, 0.f);
    for (int j = 0; j < DD; ++j) {
      const int nid = nb[j];                           // uniform -> SMEM load
      const float ind = indeg[nid];
      const float s = (ind > 0.0f) ? rsqrtf(ind) : 0.0f;
      const float4 v = *(const float4*)(feat + (size_t)nid * FF + lane * 4);
      acc.x += v.x * s; acc.y += v.y * s; acc.z += v.z * s; acc.w += v.w * s;
    }
    *(float4*)(lds + r * LSTR + lane * 4) =
        make_float4(acc.x * invSqrtD, acc.y * invSqrtD,
                    acc.z * invSqrtD, acc.w * invSqrtD);
  }
  __syncthreads();
  tile_gemm_relu_store(W, lds, out + (size_t)(BB + rowBase) * EE);
}

extern "C" void kernel_launch(void* const* d_in, const int* in_sizes, int n_in,
                              void* d_out, int out_size, void* d_ws, size_t ws_size,
                              hipStream_t stream) {
  const float* feat  = (const float*)d_in[0];   // [N, F]
  const int*   adj   = (const int*)d_in[1];     // [N, D]
  const int*   nodes = (const int*)d_in[2];     // [B]
  const float* W     = (const float*)d_in[3];   // [F, E]
  float* out = (float*)d_out;                   // [B*E] then [N*E], concatenated
  float* indeg = (float*)d_ws;                  // N floats of scratch

  // 1) zero indegree accumulator
  zero_f32_kernel<<<(NN + 255) / 256, 256, 0, stream>>>(indeg, NN);
  // 2) scatter-add indegree (column sums of the neighbor mask)
  indeg_kernel<<<(NN * DD + 255) / 256, 256, 0, stream>>>(adj, indeg);
  // 3) block 1: mean-agg + GEMM + relu  -> out rows [0, B)
  batch_agg_gemm_kernel<<<BB / 16, 256, 0, stream>>>(feat, adj, nodes, W, out);
  // 4) block 2: sym-norm agg + GEMM + relu -> out rows [B, B+N)
  neigh_agg_gemm_kernel<<<NN / 16, 256, 0, stream>>>(feat, adj, indeg, W, out);
}